// GCN_87196426044065
// MI455X (gfx1250) — compile-verified
//
#include <hip/hip_runtime.h>
#include <hip/hip_bf16.h>

typedef __attribute__((ext_vector_type(2))) float v2f;
typedef __attribute__((ext_vector_type(8))) float v8f;

#define NN 100000
#define EE 1600000
#define FIN 128
#define HH 64
#define CC 10
#define GG 256

// ---------------- degree / normalization precompute ----------------

__global__ void deg_kernel(const int* __restrict__ dst, const float* __restrict__ w,
                           float* __restrict__ deg, int E) {
    int e = blockIdx.x * blockDim.x + threadIdx.x;
    if (e < E) atomicAdd(&deg[dst[e]], w[e]);
}

__global__ void dinv_kernel(const float* __restrict__ deg, float* __restrict__ dinv, int N) {
    int i = blockIdx.x * blockDim.x + threadIdx.x;
    if (i < N) {
        float d = deg[i] + 1.0f;   // + self-loop weight 1; always > 0
        dinv[i] = rsqrtf(fmaxf(d, 1e-12f));
    }
}

__global__ void norm_kernel(const int* __restrict__ src, const int* __restrict__ dst,
                            const float* __restrict__ w, const float* __restrict__ dinv,
                            float* __restrict__ norm, int E) {
    int e = blockIdx.x * blockDim.x + threadIdx.x;
    if (e < E) norm[e] = dinv[src[e]] * w[e] * dinv[dst[e]];
}

// ---------------- dense GEMM via V_WMMA_F32_16X16X4_F32 ----------------
// out[N x 64] = A[N x K] @ W[K x 64]     (no bias; bias applied post-aggregation)
// One wave computes a 16x64 output STRIP (4 accumulators): each A fragment is
// loaded once and fed to 4 WMMAs -> 4x less L2 traffic on the large operand.
// f32 WMMA fragment layout (ISA 7.12.2):
//   A (16x4): lanes 0-15 -> K=0(v0),K=1(v1); lanes 16-31 -> K=2(v0),K=3(v1); M = lane%16
//   B (4x16): same K split across lane halves; N = lane%16
//   C/D (16x16): VGPR r holds row M=r (lanes 0-15) / M=r+8 (lanes 16-31), N = lane%16

__global__ __launch_bounds__(128) void gemm_wmma_f32(
        const float* __restrict__ A, const float* __restrict__ W,
        float* __restrict__ out, int K) {
    const int lane  = threadIdx.x & 31;
    const int mtile = blockIdx.x * 4 + (threadIdx.x >> 5);   // one 16-row strip per wave
    if (mtile * 16 >= NN) return;                            // wave-uniform: EXEC all-1s

    const int row  = mtile * 16 + (lane & 15);
    const int colb = lane & 15;                              // column within a 16-wide tile
    const int kadd = (lane >> 4) * 2;                        // 0 for lanes 0-15, 2 for 16-31

    const float* arow = A + (long)row * K + kadd;
    const float* wcol = W + (long)kadd * HH + colb;

    v8f acc0 = {}, acc1 = {}, acc2 = {}, acc3 = {};
    for (int k0 = 0; k0 < K; k0 += 4) {
        // prefetch the A stream ahead (global_prefetch_b8; speculative, safe past end)
        __builtin_prefetch(arow + k0 + 32, 0, 3);

        v2f a;
        a.x = arow[k0];
        a.y = arow[k0 + 1];

        const float* wp = wcol + (long)k0 * HH;
        v2f b0, b1, b2, b3;
        b0.x = wp[0];  b0.y = wp[HH + 0];
        b1.x = wp[16]; b1.y = wp[HH + 16];
        b2.x = wp[32]; b2.y = wp[HH + 32];
        b3.x = wp[48]; b3.y = wp[HH + 48];

        acc0 = __builtin_amdgcn_wmma_f32_16x16x4_f32(false, a, false, b0, (short)0, acc0, false, false);
        acc1 = __builtin_amdgcn_wmma_f32_16x16x4_f32(false, a, false, b1, (short)0, acc1, false, false);
        acc2 = __builtin_amdgcn_wmma_f32_16x16x4_f32(false, a, false, b2, (short)0, acc2, false, false);
        acc3 = __builtin_amdgcn_wmma_f32_16x16x4_f32(false, a, false, b3, (short)0, acc3, false, false);
    }

    const int rbase = mtile * 16 + (lane >> 4) * 8;
    float* obase = out + (long)rbase * HH + colb;
#pragma unroll
    for (int r = 0; r < 8; ++r) {
        obase[(long)r * HH +  0] = acc0[r];
        obase[(long)r * HH + 16] = acc1[r];
        obase[(long)r * HH + 32] = acc2[r];
        obase[(long)r * HH + 48] = acc3[r];
    }
}

// ---------------- edge gather-scale-scatter (L2-resident atomics) ----------------
// thread = (edge, 4-feature chunk): float4 gather from hw[src], scale, 4 atomic adds.

__global__ void scatter_kernel(const int* __restrict__ src, const int* __restrict__ dst,
                               const float* __restrict__ norm, const float* __restrict__ hw,
                               float* __restrict__ acc) {
    long t = (long)blockIdx.x * blockDim.x + threadIdx.x;
    long e = t >> 4;
    int  c = (int)(t & 15) * 4;
    if (e >= EE) return;
    float nv = norm[e];
    const float4 v = *(const float4*)(hw + (long)src[e] * HH + c);
    float* a = acc + (long)dst[e] * HH + c;
    atomicAdd(a + 0, v.x * nv);
    atomicAdd(a + 1, v.y * nv);
    atomicAdd(a + 2, v.z * nv);
    atomicAdd(a + 3, v.w * nv);
}

// self-loop contribution + bias + ReLU, in place on the accumulator
__global__ void finalize_kernel(const float* __restrict__ hw, const float* __restrict__ dinv,
                                const float* __restrict__ bias, float* __restrict__ acc) {
    long t = (long)blockIdx.x * blockDim.x + threadIdx.x;
    long i = t >> 6;
    int  f = (int)(t & 63);
    if (i >= NN) return;
    float di = dinv[i];
    float v = acc[i * HH + f] + hw[i * HH + f] * di * di + bias[f];
    acc[i * HH + f] = fmaxf(v, 0.0f);
}

// ---------------- global mean pool + FC ----------------

__global__ void pool_kernel(const float* __restrict__ h, const int* __restrict__ batch,
                            float* __restrict__ pooled, float* __restrict__ cnt) {
    long t = (long)blockIdx.x * blockDim.x + threadIdx.x;
    long i = t >> 6;
    int  f = (int)(t & 63);
    if (i >= NN) return;
    int g = batch[i];
    atomicAdd(&pooled[(long)g * HH + f], h[i * HH + f]);
    if (f == 0) atomicAdd(&cnt[g], 1.0f);
}

__global__ void fc_kernel(const float* __restrict__ pooled, const float* __restrict__ cnt,
                          const float* __restrict__ Wfc, const float* __restrict__ bfc,
                          float* __restrict__ out) {
    int t = blockIdx.x * blockDim.x + threadIdx.x;
    if (t >= GG * CC) return;
    int g = t / CC, c = t % CC;
    float inv = 1.0f / fmaxf(cnt[g], 1.0f);
    float s = bfc[c];
    const float* p = pooled + (long)g * HH;
#pragma unroll 8
    for (int f = 0; f < HH; ++f) s += p[f] * inv * Wfc[f * CC + c];
    out[t] = s;
}

// ---------------- host orchestration ----------------

extern "C" void kernel_launch(void* const* d_in, const int* in_sizes, int n_in,
                              void* d_out, int out_size, void* d_ws, size_t ws_size,
                              hipStream_t stream) {
    const float* x    = (const float*)d_in[0];                 // [N, 128]
    const int*   eidx = (const int*)d_in[1];                   // [2, E]
    const float* eatt = (const float*)d_in[2];                 // [E]
    const int*   batch= (const int*)d_in[3];                   // [N]
    const float* W1   = (const float*)d_in[4];                 // [128, 64]
    const float* b1   = (const float*)d_in[5];                 // [64]
    const float* Wh   = (const float*)d_in[6];                 // [4, 64, 64]
    const float* bh   = (const float*)d_in[7];                 // [4, 64]
    const float* Wfc  = (const float*)d_in[8];                 // [64, 10]
    const float* bfc  = (const float*)d_in[9];                 // [10]
    float* out = (float*)d_out;                                // [256, 10]

    const int* src = eidx;          // edge_index[0]
    const int* dst = eidx + EE;     // edge_index[1]

    // workspace carve-up (fp32)
    char* ws = (char*)d_ws;
    float* deg    = (float*)ws;                  ws += sizeof(float) * NN;
    float* dinv   = (float*)ws;                  ws += sizeof(float) * NN;
    float* norm   = (float*)ws;                  ws += sizeof(float) * EE;
    float* hw     = (float*)ws;                  ws += sizeof(float) * (size_t)NN * HH;
    float* hbuf   = (float*)ws;                  ws += sizeof(float) * (size_t)NN * HH;
    float* pooled = (float*)ws;                  ws += sizeof(float) * GG * HH;
    float* cnt    = (float*)ws;                  ws += sizeof(float) * GG;

    // ---- normalization precompute ----
    hipMemsetAsync(deg, 0, sizeof(float) * NN, stream);
    deg_kernel<<<(EE + 255) / 256, 256, 0, stream>>>(dst, eatt, deg, EE);
    dinv_kernel<<<(NN + 255) / 256, 256, 0, stream>>>(deg, dinv, NN);
    norm_kernel<<<(EE + 255) / 256, 256, 0, stream>>>(src, dst, eatt, dinv, norm, EE);

    const int mtiles      = NN / 16;                     // 6250 strips (N % 16 == 0)
    const int gemm_blocks = (mtiles + 3) / 4;            // 4 waves (strips) per block
    const long scat_threads = (long)EE * 16;
    const int  scat_blocks  = (int)((scat_threads + 255) / 256);
    const long nf_threads   = (long)NN * HH;
    const int  nf_blocks    = (int)((nf_threads + 255) / 256);

    // ---- layer 1: x[N,128] @ W1 ----
    gemm_wmma_f32<<<gemm_blocks, 128, 0, stream>>>(x, W1, hw, FIN);
    hipMemsetAsync(hbuf, 0, sizeof(float) * (size_t)NN * HH, stream);
    scatter_kernel<<<scat_blocks, 256, 0, stream>>>(src, dst, norm, hw, hbuf);
    finalize_kernel<<<nf_blocks, 256, 0, stream>>>(hw, dinv, b1, hbuf);

    // ---- hidden layers 2..5: h[N,64] @ Wh[i] ----
    for (int l = 0; l < 4; ++l) {
        gemm_wmma_f32<<<gemm_blocks, 128, 0, stream>>>(hbuf, Wh + (size_t)l * HH * HH, hw, HH);
        hipMemsetAsync(hbuf, 0, sizeof(float) * (size_t)NN * HH, stream);
        scatter_kernel<<<scat_blocks, 256, 0, stream>>>(src, dst, norm, hw, hbuf);
        finalize_kernel<<<nf_blocks, 256, 0, stream>>>(hw, dinv, bh + (size_t)l * HH, hbuf);
    }

    // ---- global mean pool + FC ----
    hipMemsetAsync(pooled, 0, sizeof(float) * GG * HH, stream);
    hipMemsetAsync(cnt, 0, sizeof(float) * GG, stream);
    pool_kernel<<<nf_blocks, 256, 0, stream>>>(hbuf, batch, pooled, cnt);
    fc_kernel<<<(GG * CC + 255) / 256, 256, 0, stream>>>(pooled, cnt, Wfc, bfc, out);
}